// NodeApplyModule_73993696576059
// MI455X (gfx1250) — compile-verified
//
#include <hip/hip_runtime.h>

typedef __attribute__((ext_vector_type(2))) float v2f;
typedef __attribute__((ext_vector_type(8))) float v8f;

#define N_NODES 50000
#define N_EDGES 100000
#define N_REL   64
#define DIM     50
#define KPAD    52   // 13 chunks of K=4 cover D=50
#define NPAD    64   // 4 N-tiles of 16 cover D=50

// ---------------- kernel 0: zero the f32 accumulator (d_out) ----------------
__global__ void k_zero(float* __restrict__ out, int n) {
  int i = blockIdx.x * blockDim.x + threadIdx.x;
  if (i < n) out[i] = 0.0f;
}

// ---------------- kernel 1: per-relation histogram + exclusive offsets ------
__global__ void k_bin_count(const int* __restrict__ rel, int* __restrict__ offs) {
  __shared__ int scnt[N_REL];
  for (int r = threadIdx.x; r < N_REL; r += blockDim.x) scnt[r] = 0;
  __syncthreads();
  for (int e = threadIdx.x; e < N_EDGES; e += blockDim.x)
    atomicAdd(&scnt[rel[e]], 1);
  __syncthreads();
  if (threadIdx.x == 0) {
    int run = 0;
    for (int r = 0; r < N_REL; ++r) { offs[r] = run; run += scnt[r]; }
    offs[N_REL] = run;
  }
}

// ------- kernel 2: one wave per relation, order-preserving compaction -------
__global__ __launch_bounds__(32) void k_bin_fill(const int* __restrict__ rel,
                                                 const int* __restrict__ offs,
                                                 int* __restrict__ list) {
  const int r    = blockIdx.x;
  const int lane = threadIdx.x;          // wave32
  int w = offs[r];
  for (int e0 = 0; e0 < N_EDGES; e0 += 32) {
    int e = e0 + lane;
    bool m = rel[e] == r;                // E is a multiple of 32
    unsigned mask = (unsigned)__ballot(m);
    if (m) {
      int pos = __popc(mask & ((1u << lane) - 1u));
      list[w + pos] = e;
    }
    w += __popc(mask);
  }
}

// ---------------- kernel 3: per-relation tiled GEMM via f32 WMMA ------------
// grid = (workers, N_REL); block = 1 wave (32 threads)
__global__ __launch_bounds__(32) void k_rgcn_wmma(
    const float* __restrict__ h, const float* __restrict__ weight,
    const int* __restrict__ src, const int* __restrict__ dst,
    const int* __restrict__ offs, const int* __restrict__ list,
    float* __restrict__ out) {
  __shared__ float Wsh[KPAD][NPAD];   // 13.3 KB  W[r], zero padded
  __shared__ float Hsh[16][NPAD];     //  4.0 KB  16 gathered h rows
  __shared__ int   ssrc[16], sdst[16], svalid[16];

  const int r    = blockIdx.y;
  const int lane = threadIdx.x;
  const int begin = offs[r];
  const int cnt   = offs[r + 1] - begin;
  if (cnt <= 0) return;

  // Stage W[r] into LDS, zero-padded to 52x64. B element (k,n) = W[k*D + n].
  const float* Wr = weight + (size_t)r * DIM * DIM;
  for (int i = lane; i < KPAD * NPAD; i += 32) {
    int k = i >> 6, n = i & 63;
    Wsh[k][n] = (k < DIM && n < DIM) ? Wr[k * DIM + n] : 0.0f;
  }
  __syncthreads();

  const int mrow = lane & 15;   // M index (and B/C column index)
  const int half = lane >> 4;   // lane half selects K pairs / C row half

  const int ntiles = (cnt + 15) >> 4;
  for (int t = blockIdx.x; t < ntiles; t += gridDim.x) {
    if (lane < 16) {
      int ei = (t << 4) + lane;
      bool v = ei < cnt;
      int e = v ? list[begin + ei] : 0;
      svalid[lane] = v ? 1 : 0;
      ssrc[lane]   = v ? src[e] : 0;
      sdst[lane]   = v ? dst[e] : 0;
    }
    __syncthreads();

    // Gather 16 source-node feature rows (L2-resident h) into LDS.
    for (int i = lane; i < 16 * NPAD; i += 32) {
      int m = i >> 6, k = i & 63;
      Hsh[m][k] = (k < DIM && svalid[m]) ? h[(size_t)ssrc[m] * DIM + k] : 0.0f;
    }
    __syncthreads();

    v8f c0 = {}, c1 = {}, c2 = {}, c3 = {};
    for (int kc = 0; kc < 13; ++kc) {
      const int kb = kc * 4 + 2 * half;
      // A 16x4 f32: lanes 0-15 hold K=4kc..4kc+1, lanes 16-31 hold K=4kc+2..4kc+3
      v2f a;
      a.x = Hsh[mrow][kb + 0];
      a.y = Hsh[mrow][kb + 1];
      // B 4x16 f32: VGPR j holds rows K = j + 2*half, column = mrow
      {
        v2f b; b.x = Wsh[kb + 0][ 0 + mrow]; b.y = Wsh[kb + 1][ 0 + mrow];
        c0 = __builtin_amdgcn_wmma_f32_16x16x4_f32(false, a, false, b, (short)0, c0, false, false);
      }
      {
        v2f b; b.x = Wsh[kb + 0][16 + mrow]; b.y = Wsh[kb + 1][16 + mrow];
        c1 = __builtin_amdgcn_wmma_f32_16x16x4_f32(false, a, false, b, (short)0, c1, false, false);
      }
      {
        v2f b; b.x = Wsh[kb + 0][32 + mrow]; b.y = Wsh[kb + 1][32 + mrow];
        c2 = __builtin_amdgcn_wmma_f32_16x16x4_f32(false, a, false, b, (short)0, c2, false, false);
      }
      {
        v2f b; b.x = Wsh[kb + 0][48 + mrow]; b.y = Wsh[kb + 1][48 + mrow];
        c3 = __builtin_amdgcn_wmma_f32_16x16x4_f32(false, a, false, b, (short)0, c3, false, false);
      }
    }

    // C layout: VGPR j, lanes 0-15 -> M=j, lanes 16-31 -> M=j+8; N = mrow + 16*nt.
    #pragma unroll
    for (int j = 0; j < 8; ++j) {
      int m = j + 8 * half;
      if (svalid[m]) {
        float* op = out + (size_t)sdst[m] * DIM;
        atomicAdd(op +  0 + mrow, c0[j]);
        atomicAdd(op + 16 + mrow, c1[j]);
        atomicAdd(op + 32 + mrow, c2[j]);
        if (48 + mrow < DIM) atomicAdd(op + 48 + mrow, c3[j]);
      }
    }
    __syncthreads();
  }
}

// ---------------- kernel 4: bias + ReLU in place ----------------------------
__global__ void k_bias_relu(float* __restrict__ out, const float* __restrict__ bias, int n) {
  int i = blockIdx.x * blockDim.x + threadIdx.x;
  if (i < n) {
    float v = out[i] + bias[i % DIM];
    out[i] = v > 0.0f ? v : 0.0f;
  }
}

extern "C" void kernel_launch(void* const* d_in, const int* in_sizes, int n_in,
                              void* d_out, int out_size, void* d_ws, size_t ws_size,
                              hipStream_t stream) {
  const float* h      = (const float*)d_in[0];   // [N, D]
  const float* weight = (const float*)d_in[1];   // [R, D, D]
  const float* bias   = (const float*)d_in[2];   // [D]
  const int*   src    = (const int*)d_in[3];     // [E]
  const int*   dst    = (const int*)d_in[4];     // [E]
  const int*   rel    = (const int*)d_in[5];     // [E]
  float* out = (float*)d_out;                    // [N, D] f32

  int* wsi  = (int*)d_ws;
  int* offs = wsi;         // 65 ints (exclusive offsets, offs[64] == E)
  int* list = wsi + 128;   // E ints (edge ids grouped by relation)

  const int total = N_NODES * DIM;
  k_zero<<<(total + 255) / 256, 256, 0, stream>>>(out, total);
  k_bin_count<<<1, 1024, 0, stream>>>(rel, offs);
  k_bin_fill<<<N_REL, 32, 0, stream>>>(rel, offs, list);
  dim3 grid(24, N_REL);
  k_rgcn_wmma<<<grid, 32, 0, stream>>>(h, weight, src, dst, offs, list, out);
  k_bias_relu<<<(total + 255) / 256, 256, 0, stream>>>(out, bias, total);
}